// SpanClassifier_65807488909935
// MI455X (gfx1250) — compile-verified
//
#include <hip/hip_runtime.h>
#include <hip/hip_bf16.h>

typedef __attribute__((ext_vector_type(2))) float v2f;
typedef __attribute__((ext_vector_type(8))) float v8f;

#define Hh 768
#define Ss 512
#define Bb 64
#define Ll 32
#define NSn 9
#define NEGINF (-3.402823466e+38f)

// ---------------- Kernel 1: token heads (start/end logits) via WMMA f32 16x16x4 --------
// Each wave handles 16 tokens; A = tokens[16 x 4], B cols: 0=W_start, 1=W_end, rest 0.
// B fragment built branch-free: unconditional weight loads + per-lane mask blend.
__global__ __launch_bounds__(256) void k_token_heads(
    const float* __restrict__ tokens,
    const float* __restrict__ Wst, const float* __restrict__ bst,
    const float* __restrict__ Wen, const float* __restrict__ ben,
    float* __restrict__ start_out, float* __restrict__ end_out)
{
  const int lane = threadIdx.x & 31;
  const int wave = threadIdx.x >> 5;
  const int m  = lane & 15;             // A row / B col (N)
  const int k0 = (lane >> 4) << 1;      // K pair selected by lane half
  const long rowBase = ((long)blockIdx.x * 8 + wave) * 16;   // token index base
  const float* arow = tokens + (rowBase + m) * Hh + k0;
  const float* wsrow = Wst + k0;
  const float* werow = Wen + k0;

  const float s0 = (m == 0) ? 1.f : 0.f;   // lane supplies W_start column
  const float s1 = (m == 1) ? 1.f : 0.f;   // lane supplies W_end column

  v8f c = {0.f, 0.f, 0.f, 0.f, 0.f, 0.f, 0.f, 0.f};
#pragma unroll 4
  for (int h = 0; h < Hh; h += 4) {
    v2f a  = *(const v2f*)(arow + h);     // tokens[m][h+k0 .. h+k0+1]
    v2f ws = *(const v2f*)(wsrow + h);
    v2f we = *(const v2f*)(werow + h);
    v2f bf = ws * s0 + we * s1;           // branch-free column select
    c = __builtin_amdgcn_wmma_f32_16x16x4_f32(false, a, false, bf, (short)0, c,
                                              false, false);
  }
  const int half = (lane >> 4) << 3;      // C: vgpr r -> row M = r + 8*(lane>=16)
  if (m == 0) {
    float bias = bst[0];
    for (int r = 0; r < 8; ++r) start_out[rowBase + half + r] = c[r] + bias;
  } else if (m == 1) {
    float bias = ben[0];
    for (int r = 0; r < 8; ++r) end_out[rowBase + half + r] = c[r] + bias;
  }
}

// ---------------- Kernel 2: seq heads (full_seq_logits + n_spans_logits) via WMMA -----
// 12 tiles: mt in [0,4) rows of seq_rep, nt in [0,3) columns of [W_span(32)|W_nspans(9)|0].
__global__ __launch_bounds__(32) void k_seq_heads(
    const float* __restrict__ seq,
    const float* __restrict__ Wns, const float* __restrict__ bns,
    const float* __restrict__ Wsp, const float* __restrict__ bsp,
    float* __restrict__ nspans_out, float* __restrict__ fullseq_out)
{
  const int lane = threadIdx.x;
  const int m  = lane & 15;
  const int k0 = (lane >> 4) << 1;
  const int mt = blockIdx.x & 3;
  const int nt = blockIdx.x >> 2;
  const int row  = mt * 16 + m;
  const int ncol = nt * 16 + m;
  const float* arow = seq + row * Hh + k0;

  const bool isSp = (ncol < Ll);
  const bool isNs = (ncol >= Ll) && (ncol < Ll + NSn);
  const int colSp = isSp ? ncol : 0;          // clamped -> loads always in-bounds
  const int colNs = isNs ? (ncol - Ll) : 0;
  const float sSp = isSp ? 1.f : 0.f;
  const float sNs = isNs ? 1.f : 0.f;

  v8f c = {0.f, 0.f, 0.f, 0.f, 0.f, 0.f, 0.f, 0.f};
#pragma unroll 4
  for (int h = 0; h < Hh; h += 4) {
    v2f a = *(const v2f*)(arow + h);
    float b0 = Wsp[(h + k0) * Ll + colSp] * sSp + Wns[(h + k0) * NSn + colNs] * sNs;
    float b1 = Wsp[(h + k0 + 1) * Ll + colSp] * sSp + Wns[(h + k0 + 1) * NSn + colNs] * sNs;
    v2f bf; bf.x = b0; bf.y = b1;
    c = __builtin_amdgcn_wmma_f32_16x16x4_f32(false, a, false, bf, (short)0, c,
                                              false, false);
  }
  const int rbase = mt * 16 + ((lane >> 4) << 3);
  if (isSp) {
    float bias = bsp[ncol];
    for (int r = 0; r < 8; ++r) fullseq_out[(rbase + r) * Ll + ncol] = c[r] + bias;
  } else if (isNs) {
    int j = ncol - Ll;
    float bias = bns[j];
    for (int r = 0; r < 8; ++r) nspans_out[(rbase + r) * NSn + j] = c[r] + bias;
  }
}

// ---------------- Kernel 3: per-sample top-k / merge / span pooling / outputs ---------
__global__ __launch_bounds__(256) void k_spans(
    const float* __restrict__ tokens,
    const float* __restrict__ start_lg, const float* __restrict__ end_lg,
    const float* __restrict__ nspans_lg, const float* __restrict__ fullseq,
    const float* __restrict__ Wsp, const float* __restrict__ bsp,
    float* __restrict__ spans_out, float* __restrict__ soh_out,
    float* __restrict__ eoh_out, float* __restrict__ labels_out)
{
  const int b = blockIdx.x;
  const int tid = threadIdx.x;

  __shared__ float sv[Ss];
  __shared__ float rv[256];
  __shared__ int   ri[256];
  __shared__ int   topi[2][8];
  __shared__ int   g_ns, g_cnt, g_amfull;
  __shared__ int   sa[8], ea[8];
  __shared__ float vecs[8][Hh];        // 24 KB span mean vectors
  __shared__ float lgm[8][Ll];
  __shared__ float oh0[Ss], oh1[Ss];
  __shared__ float lab[Ll], spn[Ll];

  // ---- top-8 indices for starts (which=0) and ends (which=1) ----
  // sigmoid is strictly monotone, so top-k ordering on raw logits matches the
  // reference's top_k(sigmoid(x)); ties broken toward lower index (stable).
  for (int which = 0; which < 2; ++which) {
    const float* src = (which ? end_lg : start_lg) + (long)b * Ss;
    sv[tid] = src[tid];
    sv[tid + 256] = src[tid + 256];
    __syncthreads();
    for (int t = 0; t < 8; ++t) {
      float v0 = sv[tid], v1 = sv[tid + 256];
      float bv = v0; int bi = tid;
      if (v1 > v0) { bv = v1; bi = tid + 256; }
      rv[tid] = bv; ri[tid] = bi;
      __syncthreads();
      for (int s = 128; s > 0; s >>= 1) {
        if (tid < s) {
          float ov = rv[tid + s]; int oi = ri[tid + s];
          if (ov > rv[tid] || (ov == rv[tid] && oi < ri[tid])) { rv[tid] = ov; ri[tid] = oi; }
        }
        __syncthreads();
      }
      if (tid == 0) { topi[which][t] = ri[0]; sv[ri[0]] = NEGINF; }
      __syncthreads();
    }
  }

  // ---- n_spans argmax, full-row argmax, sort, greedy merge (scalar, thread 0) ----
  if (tid == 0) {
    const float* nr = nspans_lg + b * NSn;
    int am = 0; float bm = nr[0];
    for (int j = 1; j < NSn; ++j) if (nr[j] > bm) { bm = nr[j]; am = j; }
    g_ns = am;

    const float* fr = fullseq + b * Ll;
    int amf = 0; float bmf = fr[0];
    for (int l = 1; l < Ll; ++l) if (fr[l] > bmf) { bmf = fr[l]; amf = l; }
    g_amfull = amf;

    int ssl[8], esl[8];
    for (int r = 0; r < 8; ++r) {
      ssl[r] = (r < am) ? topi[0][r] : Ss;
      esl[r] = (r < am) ? topi[1][r] : Ss;
    }
    for (int i = 1; i < 8; ++i) { int v = ssl[i]; int j = i - 1;
      while (j >= 0 && ssl[j] > v) { ssl[j + 1] = ssl[j]; --j; } ssl[j + 1] = v; }
    for (int i = 1; i < 8; ++i) { int v = esl[i]; int j = i - 1;
      while (j >= 0 && esl[j] > v) { esl[j + 1] = esl[j]; --j; } esl[j + 1] = v; }

    int sal[8] = {0,0,0,0,0,0,0,0}, eal[8] = {0,0,0,0,0,0,0,0};
    int s_i = 0, e_i = 0, n_rem = am, last_end = 0, cnt = 0;
    while (n_rem > 0 && s_i < am && e_i < am) {
      int s = ssl[s_i], e = esl[e_i];
      bool take = (e > s) && (s >= last_end);
      if (take) { sal[cnt] = s; eal[cnt] = e; last_end = e; }
      bool ele = (e <= s);
      if (take || !ele) ++s_i;
      if (take || ele)  ++e_i;
      if (take) { --n_rem; ++cnt; }
    }
    for (int r = 0; r < 8; ++r) { sa[r] = sal[r]; ea[r] = eal[r]; }
    g_cnt = cnt;
  }
  __syncthreads();

  const int ns = g_ns, cnt = g_cnt;
  const bool multi = ns > 1;           // uniform across block
  const long so = (long)b * Ss;

  if (multi) {
    // span mean vectors (spans are disjoint -> total rows read <= S)
    for (int k = 0; k < cnt; ++k) {
      int s0 = sa[k], e0 = ea[k];
      int len = e0 - s0; if (len < 1) len = 1;
      float flen = (float)len;
      for (int h = tid; h < Hh; h += 256) {
        float acc = 0.f;
        const float* tp = tokens + ((long)b * Ss + s0) * Hh + h;
        for (int r = s0; r < e0; ++r) { acc += *tp; tp += Hh; }
        vecs[k][h] = acc / flen;
      }
    }
    __syncthreads();

    {
      int k = tid >> 5, l = tid & 31;
      float v = NEGINF;
      if (k < cnt) {
        float acc = bsp[l];
        for (int h = 0; h < Hh; ++h) acc += vecs[k][h] * Wsp[h * Ll + l];
        v = acc;
      }
      lgm[k][l] = v;
    }
    if (tid < Ll) lab[tid] = 0.f;
    oh0[tid] = 0.f; oh0[tid + 256] = 0.f;
    oh1[tid] = 0.f; oh1[tid + 256] = 0.f;
    __syncthreads();

    if (tid < Ll) {
      float mx = NEGINF;
      for (int k = 0; k < cnt; ++k) mx = fmaxf(mx, lgm[k][tid]);
      spn[tid] = (cnt > 0) ? mx : 0.f;
    }
    if (tid < 8 && tid < cnt) {
      int am = 0; float bm = lgm[tid][0];
      for (int l = 1; l < Ll; ++l) if (lgm[tid][l] > bm) { bm = lgm[tid][l]; am = l; }
      lab[am] = 1.f;                   // same-value writes, benign
    }
    if (tid < cnt) { oh0[sa[tid]] = 1.f; oh1[ea[tid]] = 1.f; }
    __syncthreads();

    soh_out[so + tid] = oh0[tid]; soh_out[so + tid + 256] = oh0[tid + 256];
    eoh_out[so + tid] = oh1[tid]; eoh_out[so + tid + 256] = oh1[tid + 256];
    if (tid < Ll) {
      spans_out[b * Ll + tid]  = spn[tid];
      labels_out[b * Ll + tid] = lab[tid];
    }
  } else {
    const int as0 = topi[0][0];        // == argmax(start row), tie -> lowest idx
    const int ae0 = topi[1][0];
    soh_out[so + tid]       = (tid == as0) ? 1.f : 0.f;
    soh_out[so + tid + 256] = (tid + 256 == as0) ? 1.f : 0.f;
    eoh_out[so + tid]       = (tid == ae0) ? 1.f : 0.f;
    eoh_out[so + tid + 256] = (tid + 256 == ae0) ? 1.f : 0.f;
    if (tid < Ll) {
      spans_out[b * Ll + tid]  = fullseq[b * Ll + tid];
      labels_out[b * Ll + tid] = (tid == g_amfull) ? 1.f : 0.f;
    }
  }
}

extern "C" void kernel_launch(void* const* d_in, const int* in_sizes, int n_in,
                              void* d_out, int out_size, void* d_ws, size_t ws_size,
                              hipStream_t stream) {
  const float* tokens = (const float*)d_in[0];   // [B,S,H]
  const float* seq    = (const float*)d_in[1];   // [B,H]
  const float* Wst    = (const float*)d_in[2];   // [H]
  const float* bst    = (const float*)d_in[3];   // [1]
  const float* Wen    = (const float*)d_in[4];   // [H]
  const float* ben    = (const float*)d_in[5];   // [1]
  const float* Wns    = (const float*)d_in[6];   // [H,9]
  const float* bns    = (const float*)d_in[7];   // [9]
  const float* Wsp    = (const float*)d_in[8];   // [H,32]
  const float* bsp    = (const float*)d_in[9];   // [32]

  float* out = (float*)d_out;
  float* start_o = out;                 // [B,S]   32768
  float* end_o   = out + 32768;         // [B,S]   32768
  float* spans_o = out + 65536;         // [B,L]    2048
  float* nsp_o   = out + 67584;         // [B,9]     576
  float* soh_o   = out + 68160;         // [B,S]   32768
  float* eoh_o   = out + 100928;        // [B,S]   32768
  float* lab_o   = out + 133696;        // [B,L]    2048

  float* fullseq = (float*)d_ws;        // [B,L] scratch (fully rewritten each call)

  // 2048 token tiles of 16 rows, 8 waves/block -> 256 blocks
  k_token_heads<<<256, 256, 0, stream>>>(tokens, Wst, bst, Wen, ben, start_o, end_o);
  // 4 M-tiles x 3 N-tiles
  k_seq_heads<<<12, 32, 0, stream>>>(seq, Wns, bns, Wsp, bsp, nsp_o, fullseq);
  // per-sample irregular logic
  k_spans<<<Bb, 256, 0, stream>>>(tokens, start_o, end_o, nsp_o, fullseq, Wsp, bsp,
                                  spans_o, soh_o, eoh_o, lab_o);
}